// NanAttentionLayer_26173530702271
// MI455X (gfx1250) — compile-verified
//
#include <hip/hip_runtime.h>
#include <hip/hip_bf16.h>
#include <math.h>

#define B_SZ 256
#define E_SZ 512
#define D_SZ 512
#define NT   512            // threads per block (16 waves)
#define NW   (NT / 32)      // 16 waves

typedef __attribute__((ext_vector_type(2))) float v2f;
typedef __attribute__((ext_vector_type(8))) float v8f;

// ---------------------------------------------------------------------------
// Fully fused per-batch kernel: one workgroup per batch element b.
//   1) pool-1 scores + softmax + weighted sum over xs[b]  (HBM pass + L2 pass)
//   2) q1 = tanh(W @ r + bias) via V_WMMA_F32_16X16X4_F32 (W is L2-resident)
//   3) pool-2 with query q1                                (both passes L2-hot)
// xs[b] is 1 MiB, so all re-reads after the first streaming pass come from
// the 192 MB L2 -> effective HBM traffic ~= one pass over xs (~11.5 us).
// ---------------------------------------------------------------------------
__global__ void __launch_bounds__(NT)
fused_attn_kernel(const float* __restrict__ xs,
                  const float* __restrict__ q,
                  const float* __restrict__ W,
                  const float* __restrict__ bias,
                  float* __restrict__ out) {
    __shared__ float q_s[D_SZ];    // current pool query
    __shared__ float sc[E_SZ];     // scores -> softmax weights
    __shared__ float r_s[D_SZ];    // pool-1 result r[b]
    __shared__ float red[2 * NW];  // block-reduction scratch

    const int tid  = threadIdx.x;   // 0..511
    const int lane = tid & 31;      // wave32
    const int wid  = tid >> 5;      // 0..15
    const int b    = blockIdx.x;

    const float* xb = xs + (size_t)b * E_SZ * D_SZ;

    // ---- load shared query q into LDS ----
    q_s[tid] = q[tid];
    __syncthreads();

    // ================= POOL 1 =================
    // scores: wave w handles frames e = w, w+16, ... (coalesced over D)
    for (int e = wid; e < E_SZ; e += NW) {
        const float* row = xb + (size_t)e * D_SZ;
        float acc = 0.0f;
        #pragma unroll 4
        for (int d = lane; d < D_SZ; d += 32)
            acc = fmaf(q_s[d], row[d], acc);
        #pragma unroll
        for (int off = 16; off > 0; off >>= 1)
            acc += __shfl_xor(acc, off, 32);
        if (lane == 0) sc[e] = acc;
    }
    __syncthreads();

    // softmax over E=512 (thread owns sc[tid])
    float m = sc[tid];
    #pragma unroll
    for (int off = 16; off > 0; off >>= 1)
        m = fmaxf(m, __shfl_xor(m, off, 32));
    if (lane == 0) red[wid] = m;
    __syncthreads();
    if (tid == 0) {
        float mm = red[0];
        #pragma unroll
        for (int i = 1; i < NW; ++i) mm = fmaxf(mm, red[i]);
        red[0] = mm;
    }
    __syncthreads();
    const float gmax1 = red[0];
    float ex = __expf(sc[tid] - gmax1);
    sc[tid] = ex;
    float s = ex;
    #pragma unroll
    for (int off = 16; off > 0; off >>= 1)
        s += __shfl_xor(s, off, 32);
    if (lane == 0) red[NW + wid] = s;
    __syncthreads();
    if (tid == 0) {
        float ss = red[NW];
        #pragma unroll
        for (int i = 1; i < NW; ++i) ss += red[NW + i];
        red[NW] = 1.0f / ss;
    }
    __syncthreads();
    const float inv1 = red[NW];

    // weighted sum: thread owns d = tid; full row per iteration, coalesced,
    // xs[b] is now L2-resident.
    {
        float acc = 0.0f;
        #pragma unroll 4
        for (int e = 0; e < E_SZ; ++e)
            acc = fmaf(sc[e], xb[(size_t)e * D_SZ + tid], acc);
        r_s[tid] = acc * inv1;
    }
    __syncthreads();

    // ================= q1 = tanh(W @ r + bias) via WMMA =================
    // 32 tiles of 16 outputs; wave w handles tiles t = w, w+16.
    // B fragment: broadcast r into every column -> every output column of the
    // 16x16 tile computes the same dot(W[row,:], r); lanes 0/16 (col 0) write.
    for (int t = wid; t < D_SZ / 16; t += NW) {
        const int j0  = t * 16;
        const int row = j0 + (lane & 15);
        const int kb  = (lane >> 4) * 2;            // K sub-offset per half-wave
        const float* wrow = W + (size_t)row * D_SZ; // L2-hot (1 MiB shared)
        v8f c = {};
        for (int k = 0; k < D_SZ; k += 4) {
            v2f a = *(const v2f*)(wrow + k + kb);   // A[m, kb..kb+1] = W[row, ...]
            v2f bf;
            bf.x = r_s[k + kb];                     // B[kb,   n] = r[kb]   (all n)
            bf.y = r_s[k + kb + 1];                 // B[kb+1, n] = r[kb+1] (all n)
            c = __builtin_amdgcn_wmma_f32_16x16x4_f32(
                    false, a, false, bf, (short)0, c, false, false);
        }
        if ((lane & 15) == 0) {                     // column 0 holders
            const int half = lane >> 4;             // rows v (half 0) / v+8 (half 1)
            #pragma unroll
            for (int v = 0; v < 8; ++v) {
                const int j = j0 + v + 8 * half;
                q_s[j] = tanhf(c[v] + bias[j]);     // becomes pool-2 query
            }
        }
    }
    __syncthreads();

    // ================= POOL 2 (all xs[b] reads L2-hot) =================
    for (int e = wid; e < E_SZ; e += NW) {
        const float* row = xb + (size_t)e * D_SZ;
        float acc = 0.0f;
        #pragma unroll 4
        for (int d = lane; d < D_SZ; d += 32)
            acc = fmaf(q_s[d], row[d], acc);
        #pragma unroll
        for (int off = 16; off > 0; off >>= 1)
            acc += __shfl_xor(acc, off, 32);
        if (lane == 0) sc[e] = acc;
    }
    __syncthreads();

    m = sc[tid];
    #pragma unroll
    for (int off = 16; off > 0; off >>= 1)
        m = fmaxf(m, __shfl_xor(m, off, 32));
    if (lane == 0) red[wid] = m;
    __syncthreads();
    if (tid == 0) {
        float mm = red[0];
        #pragma unroll
        for (int i = 1; i < NW; ++i) mm = fmaxf(mm, red[i]);
        red[0] = mm;
    }
    __syncthreads();
    const float gmax2 = red[0];
    ex = __expf(sc[tid] - gmax2);
    sc[tid] = ex;
    s = ex;
    #pragma unroll
    for (int off = 16; off > 0; off >>= 1)
        s += __shfl_xor(s, off, 32);
    if (lane == 0) red[NW + wid] = s;
    __syncthreads();
    if (tid == 0) {
        float ss = red[NW];
        #pragma unroll
        for (int i = 1; i < NW; ++i) ss += red[NW + i];
        red[NW] = 1.0f / ss;
    }
    __syncthreads();
    const float inv2 = red[NW];

    {
        float acc = 0.0f;
        #pragma unroll 4
        for (int e = 0; e < E_SZ; ++e)
            acc = fmaf(sc[e], xb[(size_t)e * D_SZ + tid], acc);
        out[(size_t)b * D_SZ + tid] = acc * inv2;
    }
}

extern "C" void kernel_launch(void* const* d_in, const int* in_sizes, int n_in,
                              void* d_out, int out_size, void* d_ws, size_t ws_size,
                              hipStream_t stream) {
    const float* xs   = (const float*)d_in[0];   // [B, E, D]
    const float* q    = (const float*)d_in[1];   // [1, 1, D]
    const float* W    = (const float*)d_in[2];   // [D, D]
    const float* bias = (const float*)d_in[3];   // [D]
    float* out = (float*)d_out;                  // [B, D]

    fused_attn_kernel<<<B_SZ, NT, 0, stream>>>(xs, q, W, bias, out);
}